// GraphRCNN_2388001817051
// MI455X (gfx1250) — compile-verified
//
#include <hip/hip_runtime.h>
#include <math.h>

// ---------------- problem constants (match reference) ----------------
#define N_NODES     512
#define C_FEAT      1024
#define HID_REPN    256
#define HID         256
#define HEADS       4
#define NUM_CLASSES 91
#define KNN         5
#define N_EDGES     (N_NODES * KNN + N_NODES)   // 3072
#define REPN_LD     2052                        // repn_w1 row stride (2C+4)
#define IMG_SCALE   800.0f
#define NEG_SLOPE   0.2f

typedef __attribute__((ext_vector_type(2))) float v2f;
typedef __attribute__((ext_vector_type(8))) float v8f;

// =====================================================================
// WMMA f32 GEMM: C[M, N] = A[M, K] * B     (K multiple of 4, K >= 8)
//   BT=true : B is row-major [N,K]  (C = A * B^T)
//   BT=false: B is row-major [K,N]  (C = A * B)
//   MT      : adjacent 16-wide N tiles per wave (A fragment reused MT x)
//   NGUARD  : N not multiple of 16 -> clamp loads, select-to-zero, guard stores
// One wave computes a 16 x (16*MT) block via V_WMMA_F32_16X16X4_F32.
// Software-pipelined: fragments for step k+4 are loaded before the WMMAs
// of step k are issued, so loads overlap matrix-core latency.
// Fragment layout per CDNA5 ISA:
//   A 16x4 : lanes 0-15 M=lane (K=k0,k0+1), lanes 16-31 (K=k0+2,k0+3)
//   B 4x16 : lanes map N, VGPRs map K (same K split as A)
//   C 16x16: VGPR r -> M=r (lanes 0-15) / M=r+8 (lanes 16-31), N=lane&15
// =====================================================================
template <bool BT, int MT, bool NGUARD>
__global__ void wmma_gemm_f32(const float* __restrict__ A, int lda,
                              const float* __restrict__ B, int ldb,
                              float* __restrict__ C, int ldc,
                              int M, int N, int K) {
  const int lane    = threadIdx.x & 31;
  const int wave    = blockIdx.x * (blockDim.x >> 5) + (threadIdx.x >> 5);
  const int groupsN = (N + MT * 16 - 1) / (MT * 16);
  const int tm      = (wave / groupsN) << 4;
  const int tn0     = (wave % groupsN) * (MT * 16);
  if (tm >= M) return;

  const int row  = lane & 15;
  const int half = lane >> 4;
  const int koff = half << 1;          // lane's K sub-offset within a step

  const float* Arow = A + (tm + row) * lda;

  // per-subtile column, clamped so loads stay in-bounds
  bool nok[MT];
  const float* Bp[MT];   // BT: row pointer of B^T ; NN: column base pointer
#pragma unroll
  for (int t = 0; t < MT; ++t) {
    const int n = tn0 + t * 16 + row;
    nok[t] = NGUARD ? (n < N) : true;
    const int nc = NGUARD ? (n < N ? n : N - 1) : n;
    Bp[t] = BT ? (B + nc * ldb) : (B + nc);
  }

  const v8f vzero = {0.f, 0.f, 0.f, 0.f, 0.f, 0.f, 0.f, 0.f};
  v8f acc[MT];
#pragma unroll
  for (int t = 0; t < MT; ++t) acc[t] = vzero;

  auto loadB = [&](int t, int ka) -> v2f {
    v2f b;
    if (BT) {
      b = *(const v2f*)(Bp[t] + ka);                  // global_load_b64
    } else {
      b.x = Bp[t][ka * ldb];
      b.y = Bp[t][(ka + 1) * ldb];
    }
    if (NGUARD) {                                     // select, not branch
      b.x = nok[t] ? b.x : 0.f;
      b.y = nok[t] ? b.y : 0.f;
    }
    return b;
  };

  // prologue: fragments for k0 = 0
  v2f aC = *(const v2f*)(Arow + koff);
  v2f bC[MT];
#pragma unroll
  for (int t = 0; t < MT; ++t) bC[t] = loadB(t, koff);

#pragma unroll 2
  for (int k0 = 0; k0 < K - 4; k0 += 4) {
    // issue next step's loads first so they overlap this step's WMMAs
    const int kn = k0 + 4 + koff;
    const v2f aN = *(const v2f*)(Arow + kn);
    v2f bN[MT];
#pragma unroll
    for (int t = 0; t < MT; ++t) bN[t] = loadB(t, kn);

#pragma unroll
    for (int t = 0; t < MT; ++t)
      // (neg_a, A, neg_b, B, c_mod, C, reuse_a, reuse_b)
      acc[t] = __builtin_amdgcn_wmma_f32_16x16x4_f32(false, aC, false, bC[t],
                                                     (short)0, acc[t],
                                                     false, false);
    aC = aN;
#pragma unroll
    for (int t = 0; t < MT; ++t) bC[t] = bN[t];
  }
  // epilogue: last step
#pragma unroll
  for (int t = 0; t < MT; ++t)
    acc[t] = __builtin_amdgcn_wmma_f32_16x16x4_f32(false, aC, false, bC[t],
                                                   (short)0, acc[t],
                                                   false, false);

#pragma unroll
  for (int t = 0; t < MT; ++t) {
    if (!NGUARD || nok[t]) {
      const int n = tn0 + t * 16 + row;
      float* Cp = C + (tm + (half << 3)) * ldc + n;
#pragma unroll
      for (int r = 0; r < 8; ++r) Cp[r * ldc] = acc[t][r];
    }
  }
}

// =====================================================================
// rel[i,j] = b2 + sum_h w2[h]*relu(p1[i,h] + p2[j,h] + sum_d |bi_d-bj_d|*wg[h,d] + b1[h])
// One block per row i; p2 stored transposed (p2t[h*512+j]) for coalescing.
// =====================================================================
__global__ void rel_kernel(const float* __restrict__ p1,
                           const float* __restrict__ p2t,
                           const float* __restrict__ boxes,
                           const float* __restrict__ repn_w1,
                           const float* __restrict__ repn_b1,
                           const float* __restrict__ repn_w2,
                           const float* __restrict__ repn_b2,
                           float* __restrict__ rel) {
  __shared__ float s_p1b[HID_REPN];
  __shared__ float s_w2[HID_REPN];
  __shared__ float s_wg[4][HID_REPN];
  const int i = blockIdx.x;
  const int t = threadIdx.x;

  s_p1b[t] = p1[i * HID_REPN + t] + repn_b1[t];
  s_w2[t]  = repn_w2[t];
#pragma unroll
  for (int d = 0; d < 4; ++d)
    s_wg[d][t] = repn_w1[t * REPN_LD + 2 * C_FEAT + d];
  __syncthreads();

  float bi[4];
#pragma unroll
  for (int d = 0; d < 4; ++d) bi[d] = boxes[i * 4 + d];
  const float b2 = repn_b2[0];

  for (int j = t; j < N_NODES; j += blockDim.x) {
    float dd[4];
#pragma unroll
    for (int d = 0; d < 4; ++d) dd[d] = fabsf(bi[d] - boxes[j * 4 + d]);
    float acc = 0.f;
#pragma unroll 4
    for (int h = 0; h < HID_REPN; ++h) {
      float v = s_p1b[h] + p2t[h * N_NODES + j];
      v = fmaf(s_wg[0][h], dd[0], v);
      v = fmaf(s_wg[1][h], dd[1], v);
      v = fmaf(s_wg[2][h], dd[2], v);
      v = fmaf(s_wg[3][h], dd[3], v);
      v = v > 0.f ? v : 0.f;
      acc = fmaf(s_w2[h], v, acc);
    }
    rel[i * N_NODES + j] = acc + b2;
  }
}

// =====================================================================
// Top-(k+1) per row, drop rank-0, emit edges (src,dst) + self loops.
// Stable ties (lower index first) to match jax.lax.top_k.
// =====================================================================
__global__ void topk_edges(const float* __restrict__ rel,
                           int* __restrict__ srcE, int* __restrict__ dstE) {
  const int i = blockIdx.x * blockDim.x + threadIdx.x;
  if (i >= N_NODES) return;
  float val[KNN + 1];
  int   idx[KNN + 1];
#pragma unroll
  for (int q = 0; q <= KNN; ++q) { val[q] = -__builtin_inff(); idx[q] = 0; }
  const float* r = rel + i * N_NODES;
  for (int j = 0; j < N_NODES; ++j) {
    const float v = r[j];
    if (v > val[KNN]) {               // strict: ties keep earlier index
      int p = KNN;
      while (p > 0 && v > val[p - 1]) {
        val[p] = val[p - 1]; idx[p] = idx[p - 1]; --p;
      }
      val[p] = v; idx[p] = j;
    }
  }
#pragma unroll
  for (int q = 0; q < KNN; ++q) {
    srcE[i * KNN + q] = i;
    dstE[i * KNN + q] = idx[q + 1];   // drop top-1
  }
  srcE[N_NODES * KNN + i] = i;        // self loop
  dstE[N_NODES * KNN + i] = i;
}

// ---------------- CSR (incoming edges per dst) ----------------
__global__ void count_deg(const int* __restrict__ dstE, int* __restrict__ deg) {
  const int e = blockIdx.x * blockDim.x + threadIdx.x;
  if (e >= N_EDGES) return;
  atomicAdd(&deg[dstE[e]], 1);
}
__global__ void scan_deg(const int* __restrict__ deg, int* __restrict__ rowstart) {
  if (blockIdx.x == 0 && threadIdx.x == 0) {
    int acc = 0;
    for (int n = 0; n < N_NODES; ++n) { rowstart[n] = acc; acc += deg[n]; }
    rowstart[N_NODES] = acc;
  }
}
__global__ void fill_inc(const int* __restrict__ dstE,
                         const int* __restrict__ rowstart,
                         int* __restrict__ cursor, int* __restrict__ inc) {
  const int e = blockIdx.x * blockDim.x + threadIdx.x;
  if (e >= N_EDGES) return;
  const int d = dstE[e];
  const int p = atomicAdd(&cursor[d], 1);
  inc[rowstart[d] + p] = e;
}

// ---------------- x = [feats, geom] ----------------
__global__ void build_x(const float* __restrict__ feats,
                        const float* __restrict__ boxes,
                        float* __restrict__ x) {
  const int idx = blockIdx.x * blockDim.x + threadIdx.x;
  if (idx >= N_NODES * (C_FEAT + 4)) return;
  const int n = idx / (C_FEAT + 4);
  const int c = idx - n * (C_FEAT + 4);
  float v;
  if (c < C_FEAT) {
    v = feats[n * C_FEAT + c];
  } else {
    const float b0 = boxes[n * 4 + 0] / IMG_SCALE;
    const float b1 = boxes[n * 4 + 1] / IMG_SCALE;
    const float b2 = boxes[n * 4 + 2] / IMG_SCALE;
    const float b3 = boxes[n * 4 + 3] / IMG_SCALE;
    const int g = c - C_FEAT;
    v = (g == 0) ? b0 : (g == 1) ? b1 : (g == 2) ? (b2 - b0) : (b3 - b1);
  }
  x[idx] = v;
}

// ---------------- GAT layer 1 (4 heads x 256) ----------------
__global__ void gat1_scores(const float* __restrict__ h,
                            const float* __restrict__ asrc,
                            const float* __restrict__ adst,
                            float* __restrict__ a_s, float* __restrict__ a_d) {
  const int idx = blockIdx.x * blockDim.x + threadIdx.x;  // n*4+head
  if (idx >= N_NODES * HEADS) return;
  const int n = idx >> 2, hh = idx & 3;
  const float* hp = h + n * (HEADS * HID) + hh * HID;
  float ss = 0.f, sd = 0.f;
  for (int f = 0; f < HID; ++f) {
    const float v = hp[f];
    ss = fmaf(v, asrc[hh * HID + f], ss);
    sd = fmaf(v, adst[hh * HID + f], sd);
  }
  a_s[idx] = ss; a_d[idx] = sd;
}

__global__ void gat1_edgevals(const float* __restrict__ a_s,
                              const float* __restrict__ a_d,
                              const int* __restrict__ srcE,
                              const int* __restrict__ dstE,
                              float* __restrict__ ev) {
  const int e = blockIdx.x * blockDim.x + threadIdx.x;
  if (e >= N_EDGES) return;
  const int s = srcE[e], d = dstE[e];
#pragma unroll
  for (int hh = 0; hh < HEADS; ++hh) {
    float v = a_s[s * HEADS + hh] + a_d[d * HEADS + hh];
    ev[e * HEADS + hh] = v > 0.f ? v : NEG_SLOPE * v;
  }
}

__global__ void gat1_mdenom(const float* __restrict__ ev,
                            const int* __restrict__ rowstart,
                            const int* __restrict__ inc,
                            float* __restrict__ mOut, float* __restrict__ dOut) {
  const int idx = blockIdx.x * blockDim.x + threadIdx.x;  // n*4+head
  if (idx >= N_NODES * HEADS) return;
  const int n = idx >> 2, hh = idx & 3;
  const int s0 = rowstart[n], s1 = rowstart[n + 1];
  float m = -__builtin_inff();
  for (int p = s0; p < s1; ++p) m = fmaxf(m, ev[inc[p] * HEADS + hh]);
  if (!(m > -__builtin_inff())) m = 0.f;   // isfinite guard (empty segment)
  float s = 0.f;
  for (int p = s0; p < s1; ++p) s += expf(ev[inc[p] * HEADS + hh] - m);
  mOut[idx] = m; dOut[idx] = s;
}

__global__ void gat1_aggregate(const float* __restrict__ h,
                               const float* __restrict__ ev,
                               const float* __restrict__ mIn,
                               const float* __restrict__ dIn,
                               const int* __restrict__ srcE,
                               const int* __restrict__ rowstart,
                               const int* __restrict__ inc,
                               const float* __restrict__ bias,
                               float* __restrict__ h1) {
  const int n = blockIdx.x;
  const int t = threadIdx.x;                 // 0..255 feature within head
  const int s0 = rowstart[n], s1 = rowstart[n + 1];
  float mm[HEADS], dd[HEADS];
#pragma unroll
  for (int hh = 0; hh < HEADS; ++hh) {
    mm[hh] = mIn[n * HEADS + hh];
    dd[hh] = dIn[n * HEADS + hh] + 1e-16f;
  }
  float acc[HEADS] = {0.f, 0.f, 0.f, 0.f};
  for (int p = s0; p < s1; ++p) {
    const int e = inc[p];
    const int s = srcE[e];
    const float* hs = h + s * (HEADS * HID);
#pragma unroll
    for (int hh = 0; hh < HEADS; ++hh) {
      const float w = expf(ev[e * HEADS + hh] - mm[hh]) / dd[hh];
      acc[hh] = fmaf(w, hs[hh * HID + t], acc[hh]);
    }
  }
#pragma unroll
  for (int hh = 0; hh < HEADS; ++hh) {
    const int col = hh * HID + t;
    const float v = acc[hh] + bias[col];
    h1[n * (HEADS * HID) + col] = v > 0.f ? v : 0.f;  // relu
  }
}

// ---------------- GAT layer 2 (1 head x 91) ----------------
__global__ void gat2_scores(const float* __restrict__ z,
                            const float* __restrict__ asrc,
                            const float* __restrict__ adst,
                            float* __restrict__ a_s, float* __restrict__ a_d) {
  const int n = blockIdx.x * blockDim.x + threadIdx.x;
  if (n >= N_NODES) return;
  const float* zp = z + n * NUM_CLASSES;
  float ss = 0.f, sd = 0.f;
  for (int c = 0; c < NUM_CLASSES; ++c) {
    const float v = zp[c];
    ss = fmaf(v, asrc[c], ss);
    sd = fmaf(v, adst[c], sd);
  }
  a_s[n] = ss; a_d[n] = sd;
}

__global__ void gat2_edgevals(const float* __restrict__ a_s,
                              const float* __restrict__ a_d,
                              const int* __restrict__ srcE,
                              const int* __restrict__ dstE,
                              float* __restrict__ ev) {
  const int e = blockIdx.x * blockDim.x + threadIdx.x;
  if (e >= N_EDGES) return;
  float v = a_s[srcE[e]] + a_d[dstE[e]];
  ev[e] = v > 0.f ? v : NEG_SLOPE * v;
}

__global__ void gat2_mdenom(const float* __restrict__ ev,
                            const int* __restrict__ rowstart,
                            const int* __restrict__ inc,
                            float* __restrict__ mOut, float* __restrict__ dOut) {
  const int n = blockIdx.x * blockDim.x + threadIdx.x;
  if (n >= N_NODES) return;
  const int s0 = rowstart[n], s1 = rowstart[n + 1];
  float m = -__builtin_inff();
  for (int p = s0; p < s1; ++p) m = fmaxf(m, ev[inc[p]]);
  if (!(m > -__builtin_inff())) m = 0.f;
  float s = 0.f;
  for (int p = s0; p < s1; ++p) s += expf(ev[inc[p]] - m);
  mOut[n] = m; dOut[n] = s;
}

__global__ void gat2_aggregate(const float* __restrict__ z,
                               const float* __restrict__ ev,
                               const float* __restrict__ mIn,
                               const float* __restrict__ dIn,
                               const int* __restrict__ srcE,
                               const int* __restrict__ rowstart,
                               const int* __restrict__ inc,
                               const float* __restrict__ bias,
                               float* __restrict__ logits) {
  const int n = blockIdx.x;
  const int c = threadIdx.x;
  if (c >= NUM_CLASSES) return;
  const int s0 = rowstart[n], s1 = rowstart[n + 1];
  const float m = mIn[n], dn = dIn[n] + 1e-16f;
  float acc = 0.f;
  for (int p = s0; p < s1; ++p) {
    const int e = inc[p];
    const float w = expf(ev[e] - m) / dn;
    acc = fmaf(w, z[srcE[e] * NUM_CLASSES + c], acc);
  }
  logits[n * NUM_CLASSES + c] = acc + bias[c];
}

__global__ void softmax91(const float* __restrict__ logits,
                          float* __restrict__ probs) {
  const int n = blockIdx.x * blockDim.x + threadIdx.x;
  if (n >= N_NODES) return;
  const float* l = logits + n * NUM_CLASSES;
  float m = -__builtin_inff();
  for (int c = 0; c < NUM_CLASSES; ++c) m = fmaxf(m, l[c]);
  float s = 0.f;
  for (int c = 0; c < NUM_CLASSES; ++c) s += expf(l[c] - m);
  float* o = probs + n * NUM_CLASSES;
  for (int c = 0; c < NUM_CLASSES; ++c) o[c] = expf(l[c] - m) / s;
}

// =====================================================================
// launcher
// =====================================================================
extern "C" void kernel_launch(void* const* d_in, const int* in_sizes, int n_in,
                              void* d_out, int out_size, void* d_ws, size_t ws_size,
                              hipStream_t stream) {
  const float* feats     = (const float*)d_in[0];
  const float* boxes     = (const float*)d_in[1];
  const float* repn_w1   = (const float*)d_in[2];
  const float* repn_b1   = (const float*)d_in[3];
  const float* repn_w2   = (const float*)d_in[4];
  const float* repn_b2   = (const float*)d_in[5];
  const float* gat1_w    = (const float*)d_in[6];
  const float* gat1_asrc = (const float*)d_in[7];
  const float* gat1_adst = (const float*)d_in[8];
  const float* gat1_b    = (const float*)d_in[9];
  const float* gat2_w    = (const float*)d_in[10];
  const float* gat2_asrc = (const float*)d_in[11];
  const float* gat2_adst = (const float*)d_in[12];
  const float* gat2_b    = (const float*)d_in[13];

  float* logits = (float*)d_out;
  float* probs  = logits + (size_t)N_NODES * NUM_CLASSES;
  float* rel    = probs  + (size_t)N_NODES * NUM_CLASSES;

  // workspace carve-out (256B aligned slices)
  char* wp = (char*)d_ws;
  auto carve = [&](size_t bytes) {
    char* p = wp;
    wp += (bytes + 255) & ~(size_t)255;
    return (void*)p;
  };
  float* p1   = (float*)carve((size_t)N_NODES * HID_REPN * 4);
  float* p2t  = (float*)carve((size_t)HID_REPN * N_NODES * 4);
  float* x    = (float*)carve((size_t)N_NODES * (C_FEAT + 4) * 4);
  float* h    = (float*)carve((size_t)N_NODES * HEADS * HID * 4);
  float* h1   = (float*)carve((size_t)N_NODES * HEADS * HID * 4);
  float* z    = (float*)carve((size_t)N_NODES * NUM_CLASSES * 4);
  float* a_s1 = (float*)carve((size_t)N_NODES * HEADS * 4);
  float* a_d1 = (float*)carve((size_t)N_NODES * HEADS * 4);
  float* m1   = (float*)carve((size_t)N_NODES * HEADS * 4);
  float* dv1  = (float*)carve((size_t)N_NODES * HEADS * 4);
  float* e1   = (float*)carve((size_t)N_EDGES * HEADS * 4);
  float* a_s2 = (float*)carve((size_t)N_NODES * 4);
  float* a_d2 = (float*)carve((size_t)N_NODES * 4);
  float* m2   = (float*)carve((size_t)N_NODES * 4);
  float* dv2  = (float*)carve((size_t)N_NODES * 4);
  float* e2   = (float*)carve((size_t)N_EDGES * 4);
  int* srcE     = (int*)carve((size_t)N_EDGES * 4);
  int* dstE     = (int*)carve((size_t)N_EDGES * 4);
  int* deg      = (int*)carve((size_t)N_NODES * 4);
  int* rowstart = (int*)carve((size_t)(N_NODES + 1) * 4);
  int* cursor   = (int*)carve((size_t)N_EDGES * 4);
  int* inc      = (int*)carve((size_t)N_EDGES * 4);

  auto gemm_grid = [](int M, int N, int MT) {
    const int groupsN = (N + MT * 16 - 1) / (MT * 16);
    const int waves   = ((M + 15) / 16) * groupsN;
    return (waves + 7) / 8;  // 8 wave32s (256 threads) per block
  };

  // --- RePN: p1 = feats @ wf1^T ; p2t = wf2 @ feats^T (transposed p2) ---
  wmma_gemm_f32<true, 4, false><<<gemm_grid(N_NODES, HID_REPN, 4), 256, 0, stream>>>(
      feats, C_FEAT, repn_w1, REPN_LD, p1, HID_REPN,
      N_NODES, HID_REPN, C_FEAT);
  wmma_gemm_f32<true, 4, false><<<gemm_grid(HID_REPN, N_NODES, 4), 256, 0, stream>>>(
      repn_w1 + C_FEAT, REPN_LD, feats, C_FEAT, p2t, N_NODES,
      HID_REPN, N_NODES, C_FEAT);

  // --- pairwise relation scores (third output) ---
  rel_kernel<<<N_NODES, 256, 0, stream>>>(p1, p2t, boxes, repn_w1, repn_b1,
                                          repn_w2, repn_b2, rel);

  // --- top-k -> edges -> CSR of incoming edges ---
  topk_edges<<<(N_NODES + 255) / 256, 256, 0, stream>>>(rel, srcE, dstE);
  hipMemsetAsync(deg, 0, (size_t)N_NODES * 4, stream);
  hipMemsetAsync(cursor, 0, (size_t)N_NODES * 4, stream);
  count_deg<<<(N_EDGES + 255) / 256, 256, 0, stream>>>(dstE, deg);
  scan_deg<<<1, 1, 0, stream>>>(deg, rowstart);
  fill_inc<<<(N_EDGES + 255) / 256, 256, 0, stream>>>(dstE, rowstart, cursor, inc);

  // --- GAT input x = [feats, geom] ---
  build_x<<<(N_NODES * (C_FEAT + 4) + 255) / 256, 256, 0, stream>>>(feats, boxes, x);

  // --- GAT1: h = x @ gat1_w ---
  wmma_gemm_f32<false, 4, false><<<gemm_grid(N_NODES, HEADS * HID, 4), 256, 0, stream>>>(
      x, C_FEAT + 4, gat1_w, HEADS * HID, h, HEADS * HID,
      N_NODES, HEADS * HID, C_FEAT + 4);
  gat1_scores<<<(N_NODES * HEADS + 255) / 256, 256, 0, stream>>>(
      h, gat1_asrc, gat1_adst, a_s1, a_d1);
  gat1_edgevals<<<(N_EDGES + 255) / 256, 256, 0, stream>>>(a_s1, a_d1, srcE, dstE, e1);
  gat1_mdenom<<<(N_NODES * HEADS + 255) / 256, 256, 0, stream>>>(e1, rowstart, inc, m1, dv1);
  gat1_aggregate<<<N_NODES, HID, 0, stream>>>(h, e1, m1, dv1, srcE, rowstart, inc,
                                              gat1_b, h1);

  // --- GAT2: z = h1 @ gat2_w ---
  wmma_gemm_f32<false, 1, true><<<gemm_grid(N_NODES, NUM_CLASSES, 1), 256, 0, stream>>>(
      h1, HEADS * HID, gat2_w, NUM_CLASSES, z, NUM_CLASSES,
      N_NODES, NUM_CLASSES, HEADS * HID);
  gat2_scores<<<(N_NODES + 255) / 256, 256, 0, stream>>>(z, gat2_asrc, gat2_adst, a_s2, a_d2);
  gat2_edgevals<<<(N_EDGES + 255) / 256, 256, 0, stream>>>(a_s2, a_d2, srcE, dstE, e2);
  gat2_mdenom<<<(N_NODES + 255) / 256, 256, 0, stream>>>(e2, rowstart, inc, m2, dv2);
  gat2_aggregate<<<N_NODES, 96, 0, stream>>>(z, e2, m2, dv2, srcE, rowstart, inc,
                                             gat2_b, logits);

  // --- softmax over classes ---
  softmax91<<<(N_NODES + 255) / 256, 256, 0, stream>>>(logits, probs);
}